// BilinearSparseRouting_91036126806673
// MI455X (gfx1250) — compile-verified
//
#include <hip/hip_runtime.h>

typedef float v2f __attribute__((ext_vector_type(2)));
typedef float v8f __attribute__((ext_vector_type(8)));
typedef __attribute__((address_space(3))) float lds_float;

#define N_IN_CAPS 4096
#define B_TOTAL   256

// ---------------------------------------------------------------------------
// Phase 1: S[b] = sum_j cp4x4[b,j] @ wc[j]   via V_WMMA_F32_16X16X4_F32
//
// WMMA tile: rows M = (bb,r) = 4 batches x 4 matrix-rows; K = 4 (one capsule
// j); cols N = (bb',c) with wc[j] replicated block-diagonally. The diagonal
// 4x4 blocks of the f32 accumulator are S[b0+bb]; off-diagonals are ignored.
//
// Grid: 512 blocks x 256 threads (8 waves) = 4096 waves.
//   bgroup    = blockIdx>>3   (64 groups of 4 batches)
//   jblock    = blockIdx&7    (8 j-windows of 512 capsules)
//   wave wid  = 64-capsule chunk inside the block's j-window
//
// The block's wc slice (512 caps x 64 B = 32 KB) is staged into LDS once via
// GLOBAL_LOAD_ASYNC_TO_LDS_B64 (ASYNCcnt), then B operands come from LDS as
// paired ds reads. A operands stream straight from HBM as coalesced b64.
// Partials combine via ds_add_f32 in LDS, then global f32 atomics.
// ---------------------------------------------------------------------------
__global__ __launch_bounds__(256) void caps_phase1(
    const float* __restrict__ cp,   // (256, 4096, 16)
    const float* __restrict__ wc,   // (4096, 16) : [j*16 + k*4 + c]
    float* __restrict__ Sg)         // (64, 64) partial S, pre-zeroed
{
    __shared__ float sWc[8192];     // 512 capsules x 16 f32 = 32 KB

    int t    = threadIdx.x;
    int lane = t & 31;
    int wid  = t >> 5;
    int bgroup     = blockIdx.x >> 3;        // 0..63
    int jblockbase = (blockIdx.x & 7) << 9;  // 0,512,...,3584
    int b0 = bgroup << 2;

    // ---- async-stage this block's wc slice into LDS (32 KB) ----
    unsigned ldsBase = (unsigned)(uintptr_t)(lds_float*)sWc;  // LDS byte offset
    {
        const float* wcg = wc + ((size_t)jblockbase << 4);
#pragma unroll
        for (int i = 0; i < 16; ++i) {
            unsigned e = (unsigned)t + ((unsigned)i << 8);    // b64 element 0..4095
            unsigned ldsOff = ldsBase + (e << 3);
            unsigned long long ga =
                (unsigned long long)(uintptr_t)wcg + ((unsigned long long)e << 3);
            asm volatile("global_load_async_to_lds_b64 %0, %1, off"
                         :: "v"(ldsOff), "v"(ga) : "memory");
        }
    }
    asm volatile("s_wait_asynccnt 0x0" ::: "memory");
    __syncthreads();                 // cross-wave visibility of staged wc

    int hi = lane >> 4;              // K half: 0 -> K{0,1}, 1 -> K{2,3}
    int L  = lane & 15;
    int bb = L >> 2;                 // batch-sub (A rows) == col-block (B/D)
    int r  = L & 3;                  // matrix row (A) == matrix col c (B/D)

    int jbase = wid << 6;            // this wave's 64-capsule chunk (local)
    // A lane ptr: cp[b0+bb, j, r*4 + 2*hi + {0,1}]  (contiguous float2)
    const float* aPtr = cp
        + (((size_t)(b0 + bb) * N_IN_CAPS + jblockbase + jbase) << 4)
        + (r << 2) + (hi << 1);
    // B lane idx in LDS: sWc[jl*16 + (2*hi)*4 + c] and +4 (one ds_load_2addr)
    int bIdx = (jbase << 4) + (hi << 3) + r;

    v8f acc0 = {}; v8f acc1 = {};
    for (int it = 0; it < 32; ++it) {            // 64 capsules, 2-way unrolled
        v2f a0 = *(const v2f*)aPtr;
        v2f a1 = *(const v2f*)(aPtr + 16);
        v2f bv0; bv0.x = sWc[bIdx];      bv0.y = sWc[bIdx + 4];
        v2f bv1; bv1.x = sWc[bIdx + 16]; bv1.y = sWc[bIdx + 20];
        __builtin_prefetch(aPtr + 32 * 16, 0, 1);   // stream 32 capsules ahead
        acc0 = __builtin_amdgcn_wmma_f32_16x16x4_f32(
                   false, a0, false, bv0, (short)0, acc0, false, false);
        acc1 = __builtin_amdgcn_wmma_f32_16x16x4_f32(
                   false, a1, false, bv1, (short)0, acc1, false, false);
        aPtr += 32; bIdx += 32;
    }
    v8f acc = acc0 + acc1;

    // sWc is dead now; reuse its first 64 floats as the block reduction tile.
    __syncthreads();
    if (t < 64) sWc[t] = 0.0f;
    __syncthreads();

    // Extract diagonal blocks -> LDS. D layout: VGPR v of lane l holds
    // (M = 8*hi + v, N = l&15). Useful iff M>>2 == bb; useful v starts at
    // (bb&1)*4 within this lane's half.
    bool active = ((bb >> 1) == hi);
    bool selhi  = (bb & 1);
    float* sp = &sWc[(bb << 4) + r];     // S[bb][*, c]  (c == r variable)
#pragma unroll
    for (int rr = 0; rr < 4; ++rr) {
        float v = selhi ? acc[rr + 4] : acc[rr];
        if (active) atomicAdd(sp + (rr << 2), v);   // ds_add_f32
    }
    __syncthreads();

    if (t < 64) {
        __hip_atomic_fetch_add(&Sg[(bgroup << 6) + t], sWc[t],
                               __ATOMIC_RELAXED, __HIP_MEMORY_SCOPE_AGENT);
    }
}

// ---------------------------------------------------------------------------
// Phase 2: out[b,i] = (S[b]/32) @ wn[i]
// One WMMA per (4-batch, 4-out-capsule) tile: rows = (bb,r), cols = (ii,c),
// K = 4. A (=S) is column-independent -> every output lane is useful.
// Grid: 64 blocks (batch groups) x 8 waves (out-capsule groups of 4).
// ---------------------------------------------------------------------------
__global__ __launch_bounds__(256) void caps_phase2(
    const float* __restrict__ Sg,   // (64, 64)
    const float* __restrict__ wn,   // (32, 16) : [i*16 + k*4 + c]
    float* __restrict__ out)        // (256, 32, 16)
{
    int t = threadIdx.x;
    int lane = t & 31;
    int wid  = t >> 5;              // i-group 0..7
    int bgroup = blockIdx.x;        // 0..63
    int b0 = bgroup << 2;
    int i0 = wid << 2;

    int hi = lane >> 4;
    int L  = lane & 15;
    int bb = L >> 2;                // A: batch-sub; B/D cols: ii
    int r  = L & 3;                 // A: row;       B/D cols: c

    // A: S[b0+bb][r, 2*hi + {0,1}]  (contiguous float2)
    v2f a = *(const v2f*)(Sg + ((size_t)bgroup << 6) + (bb << 4) + (r << 2) + (hi << 1));
    // B: wn[i0+ii][2*hi + {0,1}, c]
    const float* bp = wn + ((size_t)(i0 + bb) << 4) + (hi << 3) + r;
    v2f b; b.x = bp[0]; b.y = bp[4];

    v8f c8 = {};
    c8 = __builtin_amdgcn_wmma_f32_16x16x4_f32(
             false, a, false, b, (short)0, c8, false, false);

    const float scale = 1.0f / 32.0f;       // uniform routing weights
    int colbase = ((i0 + bb) << 4) + r;     // (i*16 + c) for this lane's column
#pragma unroll
    for (int v = 0; v < 8; ++v) {
        int bbr = (hi << 1) + (v >> 2);     // row M = 8*hi+v -> batch sub-index
        int rr  = v & 3;                    // matrix row
        out[(size_t)(b0 + bbr) * 512 + colbase + (rr << 2)] = c8[v] * scale;
    }
}

extern "C" void kernel_launch(void* const* d_in, const int* in_sizes, int n_in,
                              void* d_out, int out_size, void* d_ws, size_t ws_size,
                              hipStream_t stream) {
    const float* cp = (const float*)d_in[0];   // current_pose (256,4096,16)
    const float* wc = (const float*)d_in[1];   // w_current (1,1,4096,4,4)
    const float* wn = (const float*)d_in[2];   // w_next (32,4,4)
    float* out = (float*)d_out;                // (256,1,1,32,16) fp32
    float* Sg  = (float*)d_ws;                 // 64*64 partial sums

    hipMemsetAsync(Sg, 0, 64 * 64 * sizeof(float), stream);
    caps_phase1<<<512, 256, 0, stream>>>(cp, wc, Sg);
    caps_phase2<<<64, 256, 0, stream>>>(Sg, wn, out);
}